// AbstractLayer_2001454760101
// MI455X (gfx1250) — compile-verified
//
#include <hip/hip_runtime.h>

typedef __attribute__((ext_vector_type(2))) float v2f;
typedef __attribute__((ext_vector_type(4))) float v4f;
typedef __attribute__((ext_vector_type(8))) float v8f;

#define BM 128
#define BN 128
#define BK 32
#define LDA 36    // sA row stride (floats): 16B-aligned rows + bank skew
#define LDB 132   // sB row stride (floats): 16B-aligned rows + bank skew

__device__ __forceinline__ void sign_split(float a, float& ap, float& an) {
    // Exact: an = negative part, ap = a - an (positive part). No canonicalize.
    int bits = __float_as_int(a);
    int m = bits >> 31;              // all-ones if a < 0
    an = __int_as_float(bits & m);   // a if negative else +0
    ap = a - an;                     // a if non-negative else 0
}

__device__ __forceinline__ void sign_split2(v2f a, v2f& ap, v2f& an) {
    float px, nx, py, ny;
    sign_split(a.x, px, nx);
    sign_split(a.y, py, ny);
    v2f p, n;
    p.x = px; p.y = py;
    n.x = nx; n.y = ny;
    ap = p; an = n;
}

// D = pos(A)@B1 + neg(A)@B2 for two chains selected by blockIdx.z.
// z==0: A=Wu, B1=W_upper[i], B2=W_lower[i], out=OutU
// z==1: A=Wl, B1=W_lower[i], B2=W_upper[i], out=OutL
// 4 waves / block; block tile 128x128; wave tile 64x64 (4x4 wmma tiles).
__global__ __launch_bounds__(128) void signed_gemm_kernel(
    const float* __restrict__ Wu, const float* __restrict__ Wl,
    const float* __restrict__ Bup, const float* __restrict__ Blo,
    float* __restrict__ OutU, float* __restrict__ OutL, int Dim)
{
    __shared__ float sA[BM * LDA];
    __shared__ float sB1[BK * LDB];
    __shared__ float sB2[BK * LDB];

    const int z = blockIdx.z;
    const float* __restrict__ A  = z ? Wl  : Wu;
    const float* __restrict__ B1 = z ? Blo : Bup;
    const float* __restrict__ B2 = z ? Bup : Blo;
    float* __restrict__ Cg       = z ? OutL : OutU;

    const int tid    = threadIdx.x;        // 0..127
    const int lane   = tid & 31;
    const int waveId = tid >> 5;           // 0..3
    const int waveM  = waveId & 1;         // 64-row slice
    const int waveN  = waveId >> 1;        // 64-col slice
    const int half   = lane >> 4;          // k sub-slot (+0 / +2)
    const int l16    = lane & 15;

    const int blockM = blockIdx.y * BM;
    const int blockN = blockIdx.x * BN;

    v8f acc[4][4];
#pragma unroll
    for (int mt = 0; mt < 4; ++mt)
#pragma unroll
        for (int nt = 0; nt < 4; ++nt)
            acc[mt][nt] = (v8f)(0.0f);

    // Global->LDS tile load mapping (128 threads)
    const int aCol4 = (tid & 7) * 4;   // 8 float4 per 32-wide A row
    const int aRow0 = tid >> 3;        // 0..15, 8 passes (+16)
    const int bCol4 = (tid & 31) * 4;  // 32 float4 per 128-wide B row
    const int bRow0 = tid >> 5;        // 0..3, 8 passes (+4)

    for (int k0 = 0; k0 < Dim; k0 += BK) {
        __syncthreads();
#pragma unroll
        for (int p = 0; p < 8; ++p) {
            int r = aRow0 + p * 16;
            v4f v = *(const v4f*)(A + (size_t)(blockM + r) * Dim + k0 + aCol4);
            *(v4f*)(sA + r * LDA + aCol4) = v;
        }
#pragma unroll
        for (int p = 0; p < 8; ++p) {
            int r = bRow0 + p * 4;
            v4f v1 = *(const v4f*)(B1 + (size_t)(k0 + r) * Dim + blockN + bCol4);
            v4f v2 = *(const v4f*)(B2 + (size_t)(k0 + r) * Dim + blockN + bCol4);
            *(v4f*)(sB1 + r * LDB + bCol4) = v1;
            *(v4f*)(sB2 + r * LDB + bCol4) = v2;
        }
        __syncthreads();

        // Prefetch next k-tile into cache while this tile computes.
        if (k0 + BK < Dim) {
            int kn = k0 + BK;
            __builtin_prefetch(A + (size_t)(blockM + tid) * Dim + kn, 0, 3);
            __builtin_prefetch(B1 + (size_t)(kn + (tid >> 2)) * Dim + blockN + (tid & 3) * 32, 0, 3);
            __builtin_prefetch(B2 + (size_t)(kn + (tid >> 2)) * Dim + blockN + (tid & 3) * 32, 0, 3);
        }

        // Limit unroll to keep VGPR allocation under 256 (avoid MSB switching);
        // 32 multi-cycle WMMAs per step already hide LDS latency.
#pragma unroll 2
        for (int kk = 0; kk < BK; kk += 4) {
            const int ka = kk + 2 * half;
            v2f ap[4], an[4];
#pragma unroll
            for (int mt = 0; mt < 4; ++mt) {
                int m = waveM * 64 + mt * 16 + l16;
                v2f av = *(const v2f*)(sA + m * LDA + ka);   // ds b64
                sign_split2(av, ap[mt], an[mt]);
            }
#pragma unroll
            for (int nt = 0; nt < 4; ++nt) {
                int n = waveN * 64 + nt * 16 + l16;
                v2f b1f; b1f.x = sB1[ka * LDB + n]; b1f.y = sB1[(ka + 1) * LDB + n];
                v2f b2f; b2f.x = sB2[ka * LDB + n]; b2f.y = sB2[(ka + 1) * LDB + n];
#pragma unroll
                for (int mt = 0; mt < 4; ++mt) {
                    acc[mt][nt] = __builtin_amdgcn_wmma_f32_16x16x4_f32(
                        false, ap[mt], false, b1f, (short)0, acc[mt][nt], false, false);
                    acc[mt][nt] = __builtin_amdgcn_wmma_f32_16x16x4_f32(
                        false, an[mt], false, b2f, (short)0, acc[mt][nt], false, false);
                }
            }
        }
    }

    // Epilogue: C layout -> VGPR j holds row (j + 8*half), col = l16
#pragma unroll
    for (int mt = 0; mt < 4; ++mt) {
#pragma unroll
        for (int nt = 0; nt < 4; ++nt) {
            int col = blockN + waveN * 64 + nt * 16 + l16;
            int row = blockM + waveM * 64 + mt * 16 + 8 * half;
#pragma unroll
            for (int j = 0; j < 8; ++j)
                Cg[(size_t)(row + j) * Dim + col] = acc[mt][nt][j];
        }
    }
}

// One wave per row: computes concrete bounds (min/max against best) and,
// if update!=0, the bias back-substitution for both chains in one pass.
__global__ __launch_bounds__(256) void bounds_kernel(
    const float* __restrict__ Wu, const float* __restrict__ Wl,
    const float* __restrict__ ubi, const float* __restrict__ lbi,
    const float* __restrict__ bUi, const float* __restrict__ bLi,
    float* __restrict__ bu, float* __restrict__ bl,
    const float* __restrict__ bestU, const float* __restrict__ bestL,
    float* __restrict__ outU, float* __restrict__ outL,
    int update, int Dim)
{
    const int wave = threadIdx.x >> 5;
    const int lane = threadIdx.x & 31;
    const int row  = blockIdx.x * 8 + wave;
    if (row >= Dim) return;

    const float* __restrict__ wuR = Wu + (size_t)row * Dim;
    const float* __restrict__ wlR = Wl + (size_t)row * Dim;

    float su_c = 0.0f, sl_c = 0.0f, su_b = 0.0f, sl_b = 0.0f;
    for (int k = lane; k < Dim; k += 32) {
        float wu = wuR[k], wl = wlR[k];
        float u = ubi[k], l = lbi[k];
        float wup, wun, wlp, wln;
        sign_split(wu, wup, wun);
        sign_split(wl, wlp, wln);
        su_c += wup * u + wun * l;
        sl_c += wlp * l + wln * u;
        if (update) {
            float bU = bUi[k], bL = bLi[k];
            su_b += wup * bU + wun * bL;
            sl_b += wlp * bL + wln * bU;
        }
    }
#pragma unroll
    for (int off = 16; off > 0; off >>= 1) {
        su_c += __shfl_xor(su_c, off, 32);
        sl_c += __shfl_xor(sl_c, off, 32);
        su_b += __shfl_xor(su_b, off, 32);
        sl_b += __shfl_xor(sl_b, off, 32);
    }
    if (lane == 0) {
        float bue = bu[row], ble = bl[row];
        outU[row] = fminf(bestU[row], su_c + bue);
        outL[row] = fmaxf(bestL[row], sl_c + ble);
        if (update) {
            bu[row] = su_b + bue;
            bl[row] = sl_b + ble;
        }
    }
}

extern "C" void kernel_launch(void* const* d_in, const int* in_sizes, int n_in,
                              void* d_out, int out_size, void* d_ws, size_t ws_size,
                              hipStream_t stream) {
    const float* W_upper = (const float*)d_in[0];
    const float* W_lower = (const float*)d_in[1];
    const float* b_upper = (const float*)d_in[2];
    const float* b_lower = (const float*)d_in[3];
    const float* ub      = (const float*)d_in[4];
    const float* lb      = (const float*)d_in[5];
    float* out = (float*)d_out;

    const int L = 6;
    const int Dim = in_sizes[4] / L;   // 2048
    const size_t msz = (size_t)Dim * Dim;

    float* Wu_a  = (float*)d_ws;
    float* Wl_a  = Wu_a + msz;
    float* Wu_b  = Wl_a + msz;
    float* Wl_b  = Wu_b + msz;
    float* bu    = Wl_b + msz;
    float* bl    = bu + Dim;
    float* bestU = bl + Dim;
    float* bestL = bestU + Dim;

    // Initialize state from layer L-1 (graph-capture-safe d2d copies)
    (void)hipMemcpyAsync(Wu_a, W_upper + (size_t)(L - 1) * msz, msz * sizeof(float),
                         hipMemcpyDeviceToDevice, stream);
    (void)hipMemcpyAsync(Wl_a, W_lower + (size_t)(L - 1) * msz, msz * sizeof(float),
                         hipMemcpyDeviceToDevice, stream);
    (void)hipMemcpyAsync(bu, b_upper + (size_t)(L - 1) * Dim, Dim * sizeof(float),
                         hipMemcpyDeviceToDevice, stream);
    (void)hipMemcpyAsync(bl, b_lower + (size_t)(L - 1) * Dim, Dim * sizeof(float),
                         hipMemcpyDeviceToDevice, stream);
    (void)hipMemcpyAsync(bestU, ub + (size_t)(L - 1) * Dim, Dim * sizeof(float),
                         hipMemcpyDeviceToDevice, stream);
    (void)hipMemcpyAsync(bestL, lb + (size_t)(L - 1) * Dim, Dim * sizeof(float),
                         hipMemcpyDeviceToDevice, stream);

    float* Wu_cur = Wu_a; float* Wl_cur = Wl_a;
    float* Wu_nxt = Wu_b; float* Wl_nxt = Wl_b;

    const dim3 gB(Dim / 8, 1, 1);
    const dim3 gG(Dim / BN, Dim / BM, 2);

    for (int i = L - 2; i >= 0; --i) {
        // concrete bounds + bias back-substitution (reads Wu/Wl before update)
        bounds_kernel<<<gB, 256, 0, stream>>>(
            Wu_cur, Wl_cur,
            ub + (size_t)i * Dim, lb + (size_t)i * Dim,
            b_upper + (size_t)i * Dim, b_lower + (size_t)i * Dim,
            bu, bl, bestU, bestL, bestU, bestL, 1, Dim);
        // symbolic matrix composition
        signed_gemm_kernel<<<gG, 128, 0, stream>>>(
            Wu_cur, Wl_cur,
            W_upper + (size_t)i * msz, W_lower + (size_t)i * msz,
            Wu_nxt, Wl_nxt, Dim);
        float* t;
        t = Wu_cur; Wu_cur = Wu_nxt; Wu_nxt = t;
        t = Wl_cur; Wl_cur = Wl_nxt; Wl_nxt = t;
    }

    // Final concretization against layer-0 input bounds -> d_out
    bounds_kernel<<<gB, 256, 0, stream>>>(
        Wu_cur, Wl_cur, ub, lb, nullptr, nullptr,
        bu, bl, bestU, bestL, out, out + Dim, 0, Dim);
}